// MultipoleFieldRefiner_80805514707454
// MI455X (gfx1250) — compile-verified
//
#include <hip/hip_runtime.h>
#include <hip/hip_bf16.h>
#include <math.h>

// ---------------------------------------------------------------------------
// MultipoleFieldRefiner on MI455X (gfx1250, wave32, WMMA)
// B=16, N=512, D=128, H=256.
//  * MLP GEMMs: v_wmma_f32_16x16x4_f32, one 16x64 macro-tile per wave
//    (A fragment reused across 4 B fragments -> 4 WMMA per A load).
//  * Field kernel: j-atom tile staged into LDS with
//    GLOBAL_LOAD_ASYNC_TO_LDS_B32 (ASYNCcnt) -- the CDNA5 async path.
// ---------------------------------------------------------------------------

#define BB 16
#define NN 512
#define DD 128
#define HH 256
#define NTOK (BB * NN)          // 8192
#define CUTOFF 5.0f
#define SIGMA 0.6f
#define EPSF 0.15f
#define PI_F 3.14159265358979323846f

typedef float v2f __attribute__((ext_vector_type(2)));
typedef float v8f __attribute__((ext_vector_type(8)));

// ---- CDNA5 async global->LDS copy (no builtin; inline asm, GVS mode) ------
__device__ __forceinline__ void async_g2l_b32(void* lds_ptr, const void* gbase,
                                              unsigned byte_off) {
    // LDS aperture: low 32 bits of a generic shared pointer == LDS byte offset.
    unsigned lds_off = (unsigned)(uintptr_t)lds_ptr;
    asm volatile("global_load_async_to_lds_b32 %0, %1, %2"
                 :: "v"(lds_off), "v"(byte_off), "s"(gbase)
                 : "memory");
}
__device__ __forceinline__ void wait_async0() {
    asm volatile("s_wait_asynccnt 0x0" ::: "memory");
}

// ---------------------------------------------------------------------------
// Wave-tile GEMM:  C[M,N] = act(A[M,K] @ W[K,N] + bias[N])
// One wave computes a 16 x (16*NPW) macro-tile; K stepped by 4 via
// V_WMMA_F32_16X16X4_F32.  M,N multiples of 16; K multiple of 4.
// act: 0=linear, 1=SiLU.
// ---------------------------------------------------------------------------
template <int NPW>
__global__ void gemm_wmma_f32(const float* __restrict__ A, int lda,
                              const float* __restrict__ W, int ldw,
                              const float* __restrict__ bias,
                              float* __restrict__ C, int ldc,
                              int M, int N, int K, int act) {
    int wave    = blockIdx.x * (blockDim.x >> 5) + (threadIdx.x >> 5);
    int groupsN = (N >> 4) / NPW;
    int nTiles  = (M >> 4) * groupsN;
    if (wave >= nTiles) return;               // wave-uniform skip
    int tm  = (wave / groupsN) << 4;
    int tn0 = (wave % groupsN) * (NPW << 4);

    int lane = threadIdx.x & 31;
    int half = lane >> 4;                     // 0: K=k,k+1   1: K=k+2,k+3
    int l16  = lane & 15;

    const float* Arow = A + (size_t)(tm + l16) * lda;

    v8f acc[NPW];
#pragma unroll
    for (int t = 0; t < NPW; ++t) acc[t] = {};

    for (int k = 0; k < K; k += 4) {
        int kk = k + (half << 1);
        v2f a = *reinterpret_cast<const v2f*>(Arow + kk);   // 8B aligned
#pragma unroll
        for (int t = 0; t < NPW; ++t) {
            const float* Bc = W + tn0 + (t << 4) + l16;
            v2f b;
            b.x = Bc[(size_t)kk * ldw];
            b.y = Bc[(size_t)(kk + 1) * ldw];
            acc[t] = __builtin_amdgcn_wmma_f32_16x16x4_f32(
                false, a, false, b, (short)0, acc[t], false, false);
        }
    }

#pragma unroll
    for (int t = 0; t < NPW; ++t) {
        int col = tn0 + (t << 4) + l16;
        float bv = bias[col];
#pragma unroll
        for (int r = 0; r < 8; ++r) {
            int row = tm + r + (half << 3);   // C layout: VGPR r -> M=r / r+8
            float v = acc[t][r] + bv;
            if (act) v = v / (1.0f + __expf(-v));  // SiLU
            C[(size_t)row * ldc + col] = v;
        }
    }
}

// ---------------------------------------------------------------------------
// Zero-padded weight / bias copy:  dst[dr,dc] <- src[sr,sc] (zero elsewhere)
// ---------------------------------------------------------------------------
__global__ void pad_copy_kernel(const float* __restrict__ src, int sr, int sc,
                                float* __restrict__ dst, int dr, int dc) {
    int g = blockIdx.x * blockDim.x + threadIdx.x;
    if (g >= dr * dc) return;
    int r = g / dc, c = g % dc;
    dst[g] = (r < sr && c < sc) ? src[r * sc + c] : 0.0f;
}

// X0 = [h | 0 | 0 | pad]  -> [NTOK, 144]
__global__ void build_x0_kernel(const float* __restrict__ h, float* __restrict__ X) {
    int g = blockIdx.x * blockDim.x + threadIdx.x;
    if (g >= NTOK * 144) return;
    int row = g / 144, col = g % 144;
    X[g] = (col < DD) ? h[row * DD + col] : 0.0f;
}

// Updater input: [h | q | m | mu_q | mu_m | vq | vm | eq | em] -> [NTOK,144]
__global__ void build_xu_kernel(const float* __restrict__ h,
                                const float* __restrict__ q, const float* __restrict__ m,
                                const float* __restrict__ muq, const float* __restrict__ mum,
                                const float* __restrict__ vq, const float* __restrict__ vm,
                                const float* __restrict__ eq, const float* __restrict__ em,
                                float* __restrict__ X) {
    int g = blockIdx.x * blockDim.x + threadIdx.x;
    if (g >= NTOK * 144) return;
    int row = g / 144, col = g % 144;
    float v;
    if      (col < 128) v = h[row * DD + col];
    else if (col == 128) v = q[row];
    else if (col == 129) v = m[row];
    else if (col < 133)  v = muq[row * 3 + (col - 130)];
    else if (col < 136)  v = mum[row * 3 + (col - 133)];
    else if (col == 136) v = vq[row];
    else if (col == 137) v = vm[row];
    else if (col < 141)  v = eq[row * 3 + (col - 138)];
    else                 v = em[row * 3 + (col - 141)];
    X[g] = v;
}

__device__ __forceinline__ float gather16(int idx, int col,
        const float* q, const float* m, const float* muq, const float* mum,
        const float* vq, const float* vm, const float* eq, const float* em) {
    if (col == 0) return q[idx];
    if (col == 1) return m[idx];
    if (col < 5)  return muq[idx * 3 + (col - 2)];
    if (col < 8)  return mum[idx * 3 + (col - 5)];
    if (col == 8) return vq[idx];
    if (col == 9) return vm[idx];
    if (col < 13) return eq[idx * 3 + (col - 10)];
    return em[idx * 3 + (col - 13)];
}

__global__ void build_atom_in_kernel(const float* q, const float* m,
                                     const float* muq, const float* mum,
                                     const float* vq, const float* vm,
                                     const float* eq, const float* em,
                                     float* __restrict__ out) {
    int g = blockIdx.x * blockDim.x + threadIdx.x;
    if (g >= NTOK * 16) return;
    out[g] = gather16(g / 16, g % 16, q, m, muq, mum, vq, vm, eq, em);
}

__global__ void build_abs_in_kernel(const float* q, const float* m,
                                    const float* muq, const float* mum,
                                    const float* vq, const float* vm,
                                    const float* eq, const float* em,
                                    const int* __restrict__ absIdx,
                                    float* __restrict__ out) {
    int t = blockIdx.x * blockDim.x + threadIdx.x;
    if (t >= BB * 16) return;
    int b = t >> 4, col = t & 15;
    int a = absIdx[0];
    out[t] = gather16(b * NN + a, col, q, m, muq, mum, vq, vm, eq, em);
}

// ---------------------------------------------------------------------------
// Equilibration (mask is all-true in this problem).
// ---------------------------------------------------------------------------
__global__ void equil_init_kernel(const float* __restrict__ U,   // [NTOK,16]
                                  float* q, float* m, float* muq, float* mum) {
    __shared__ float sA[NN], sB[NN];
    int b = blockIdx.x, tid = threadIdx.x;
    int idx = b * NN + tid;
    const float* u = U + (size_t)idx * 16;
    float qi = u[0], mi = u[1];
    muq[idx * 3 + 0] = u[2]; muq[idx * 3 + 1] = u[3]; muq[idx * 3 + 2] = u[4];
    mum[idx * 3 + 0] = u[5]; mum[idx * 3 + 1] = u[6]; mum[idx * 3 + 2] = u[7];
    sA[tid] = qi; sB[tid] = mi;
    __syncthreads();
    for (int s = NN / 2; s > 0; s >>= 1) {
        if (tid < s) { sA[tid] += sA[tid + s]; sB[tid] += sB[tid + s]; }
        __syncthreads();
    }
    float invN = 1.0f / (float)NN;
    q[idx] = qi - sA[0] * invN;
    m[idx] = mi - sB[0] * invN;
}

__device__ __forceinline__ float softplusf(float x) {
    return fmaxf(x, 0.0f) + log1pf(__expf(-fabsf(x)));
}
__device__ __forceinline__ float sigmoidf(float x) {
    return 1.0f / (1.0f + __expf(-x));
}

__global__ void update_equil_kernel(const float* __restrict__ U,  // [NTOK,16]
                                    float* q, float* m, float* muq, float* mum) {
    __shared__ float sA[NN], sB[NN], sC[NN], sD[NN];
    int b = blockIdx.x, tid = threadIdx.x;
    int idx = b * NN + tid;
    const float* u = U + (size_t)idx * 16;
    float qi  = q[idx] + u[0];
    float mi  = m[idx] + u[1];
    float fqi = softplusf(u[8]) + 1e-6f;
    float fmi = softplusf(u[9]) + 1e-6f;
    float aq  = sigmoidf(u[10]);
    float am  = sigmoidf(u[11]);
#pragma unroll
    for (int c = 0; c < 3; ++c) {
        muq[idx * 3 + c] += aq * u[2 + c];
        mum[idx * 3 + c] += am * u[5 + c];
    }
    sA[tid] = qi; sB[tid] = mi; sC[tid] = fqi; sD[tid] = fmi;
    __syncthreads();
    for (int s = NN / 2; s > 0; s >>= 1) {
        if (tid < s) {
            sA[tid] += sA[tid + s]; sB[tid] += sB[tid + s];
            sC[tid] += sC[tid + s]; sD[tid] += sD[tid + s];
        }
        __syncthreads();
    }
    q[idx] = qi - fqi / fmaxf(sC[0], 1e-8f) * sA[0];
    m[idx] = mi - fmi / fmaxf(sD[0], 1e-8f) * sB[0];
}

// ---------------------------------------------------------------------------
// Pairwise multipole field.  grid = (N/256, B), block = 256.
// All 512 j-atoms (pos,q,m,mu_q,mu_m = 22 KB) staged into LDS with
// GLOBAL_LOAD_ASYNC_TO_LDS_B32; completion via s_wait_asynccnt + barrier.
// ---------------------------------------------------------------------------
__global__ void field_kernel(const float* __restrict__ pos,
                             const float* __restrict__ q, const float* __restrict__ m,
                             const float* __restrict__ muq, const float* __restrict__ mum,
                             float* __restrict__ vq, float* __restrict__ vm,
                             float* __restrict__ eq, float* __restrict__ em) {
    __shared__ float smem[NN * 11];
    float* sp  = smem;            // NN*3
    float* sq  = smem + NN * 3;   // NN
    float* sm  = smem + NN * 4;   // NN
    float* sdq = smem + NN * 5;   // NN*3
    float* sdm = smem + NN * 8;   // NN*3

    int b = blockIdx.y, tid = threadIdx.x;
    int base = b * NN;

    for (int j = tid; j < NN; j += blockDim.x) {
        unsigned e  = (unsigned)(base + j);
#pragma unroll
        for (int c = 0; c < 3; ++c) {
            async_g2l_b32(&sp[j * 3 + c],  pos, (e * 3 + c) * 4u);
            async_g2l_b32(&sdq[j * 3 + c], muq, (e * 3 + c) * 4u);
            async_g2l_b32(&sdm[j * 3 + c], mum, (e * 3 + c) * 4u);
        }
        async_g2l_b32(&sq[j], q, e * 4u);
        async_g2l_b32(&sm[j], m, e * 4u);
    }
    wait_async0();
    __syncthreads();

    int i = blockIdx.x * blockDim.x + tid;
    float px = sp[i * 3 + 0], py = sp[i * 3 + 1], pz = sp[i * 3 + 2];
    float avq = 0.f, avm = 0.f;
    float eqx = 0.f, eqy = 0.f, eqz = 0.f;
    float emx = 0.f, emy = 0.f, emz = 0.f;

    for (int j = 0; j < NN; ++j) {
        float rx = px - sp[j * 3 + 0];
        float ry = py - sp[j * 3 + 1];
        float rz = pz - sp[j * 3 + 2];
        float d2  = rx * rx + ry * ry + rz * rz;
        float dij = sqrtf(fmaxf(d2, 1e-16f));
        float pairf = (dij > 1e-8f && dij <= CUTOFF) ? 1.0f : 0.0f;
        float invd = 1.0f / fmaxf(dij, 1e-8f);
        float ux = rx * invd, uy = ry * invd, uz = rz * invd;
        float invr = rsqrtf(d2 + EPSF * EPSF);
        float s = dij / SIGMA;
        float w = 0.5f * (__cosf(PI_F * dij / CUTOFF) + 1.0f) * pairf
                  * __expf(-0.5f * s * s);
        float wi  = w * invr;
        float wi2 = wi * invr;
        float wi3 = wi2 * invr;
        float qj = sq[j], mj = sm[j];
        float mqd = sdq[j * 3 + 0] * ux + sdq[j * 3 + 1] * uy + sdq[j * 3 + 2] * uz;
        float mmd = sdm[j * 3 + 0] * ux + sdm[j * 3 + 1] * uy + sdm[j * 3 + 2] * uz;
        avq += wi * qj + wi2 * mqd;
        avm += wi * mj + wi2 * mmd;
        float cq = wi2 * qj + wi3 * mqd;
        float cm = wi2 * mj + wi3 * mmd;
        eqx += cq * ux; eqy += cq * uy; eqz += cq * uz;
        emx += cm * ux; emy += cm * uy; emz += cm * uz;
    }
    vq[base + i] = avq;
    vm[base + i] = avm;
    eq[(size_t)(base + i) * 3 + 0] = eqx;
    eq[(size_t)(base + i) * 3 + 1] = eqy;
    eq[(size_t)(base + i) * 3 + 2] = eqz;
    em[(size_t)(base + i) * 3 + 0] = emx;
    em[(size_t)(base + i) * 3 + 1] = emy;
    em[(size_t)(base + i) * 3 + 2] = emz;
}

// Pack q,m,mu_q,mu_m,vq,vm,eq,em into d_out (first 131072 floats, return order)
__global__ void copy_outputs_kernel(const float* q, const float* m,
                                    const float* muq, const float* mum,
                                    const float* vq, const float* vm,
                                    const float* eq, const float* em,
                                    float* __restrict__ out) {
    int g = blockIdx.x * blockDim.x + threadIdx.x;
    if (g >= NTOK * 16) return;
    float v;
    if      (g < 8192)   v = q[g];
    else if (g < 16384)  v = m[g - 8192];
    else if (g < 40960)  v = muq[g - 16384];
    else if (g < 65536)  v = mum[g - 40960];
    else if (g < 73728)  v = vq[g - 65536];
    else if (g < 81920)  v = vm[g - 73728];
    else if (g < 106496) v = eq[g - 81920];
    else                 v = em[g - 106496];
    out[g] = v;
}

// ---------------------------------------------------------------------------
// Host orchestration
// ---------------------------------------------------------------------------
extern "C" void kernel_launch(void* const* d_in, const int* in_sizes, int n_in,
                              void* d_out, int out_size, void* d_ws, size_t ws_size,
                              hipStream_t stream) {
    const float* h      = (const float*)d_in[0];
    const float* pos    = (const float*)d_in[1];
    const int*   absIdx = (const int*)d_in[3];

    // Param group bases: insertion order (init,upd,atom,abs) or jax-sorted
    // (abs,atom,init,upd); disambiguate via W0 size (init W0 = 130*256).
    int I_init = 4, I_upd = 10, I_atom = 16, I_abs = 22;
    if (in_sizes[4] != 130 * 256) { I_abs = 4; I_atom = 10; I_init = 16; I_upd = 22; }
    const float* iW0 = (const float*)d_in[I_init + 0];
    const float* ib0 = (const float*)d_in[I_init + 1];
    const float* iW1 = (const float*)d_in[I_init + 2];
    const float* ib1 = (const float*)d_in[I_init + 3];
    const float* iW2 = (const float*)d_in[I_init + 4];
    const float* ib2 = (const float*)d_in[I_init + 5];
    const float* uW0 = (const float*)d_in[I_upd + 0];
    const float* ub0 = (const float*)d_in[I_upd + 1];
    const float* uW1 = (const float*)d_in[I_upd + 2];
    const float* ub1 = (const float*)d_in[I_upd + 3];
    const float* uW2 = (const float*)d_in[I_upd + 4];
    const float* ub2 = (const float*)d_in[I_upd + 5];
    const float* aW0 = (const float*)d_in[I_atom + 0];
    const float* ab0 = (const float*)d_in[I_atom + 1];
    const float* aW1 = (const float*)d_in[I_atom + 2];
    const float* ab1 = (const float*)d_in[I_atom + 3];
    const float* aW2 = (const float*)d_in[I_atom + 4];
    const float* ab2 = (const float*)d_in[I_atom + 5];
    const float* sW0 = (const float*)d_in[I_abs + 0];
    const float* sb0 = (const float*)d_in[I_abs + 1];
    const float* sW1 = (const float*)d_in[I_abs + 2];
    const float* sb1 = (const float*)d_in[I_abs + 3];
    const float* sW2 = (const float*)d_in[I_abs + 4];
    const float* sb2 = (const float*)d_in[I_abs + 5];

    float* ws = (float*)d_ws;
    float* X     = ws;                       // 8192*144
    float* H1    = ws + 1179648;             // 8192*256
    float* H2    = ws + 3276800;             // 8192*256
    float* U16   = ws + 5373952;             // 8192*16
    float* q     = ws + 5505024;             // 8192
    float* m     = ws + 5513216;             // 8192
    float* muq   = ws + 5521408;             // 24576
    float* mum   = ws + 5545984;             // 24576
    float* vq    = ws + 5570560;             // 8192
    float* vm    = ws + 5578752;             // 8192
    float* eqb   = ws + 5586944;             // 24576
    float* emb   = ws + 5611520;             // 24576
    float* Wp144 = ws + 5636096;             // 144*256 (init W0 padded)
    float* WpI2  = ws + 5672960;             // 256*16  (init W2 padded)
    float* bpI2  = ws + 5677056;             // 16
    float* WpU2  = ws + 5677072;             // 256*16  (upd  W2 padded)
    float* bpU2  = ws + 5681168;             // 16
    float* absin = ws + 5681184;             // 16*16

    float* out      = (float*)d_out;
    float* out_atom = out + 131072;          // field_atom_lat [8192,256]
    float* out_abs  = out + 2228224;         // field_abs_feat [16,256]

    auto ge = [&](const float* A, int lda, const float* W, int ldw, const float* bias,
                  float* C, int ldc, int M, int Nn, int K, int act) {
        if ((Nn & 63) == 0) {
            int nTiles = (M / 16) * (Nn / 64);
            int blocks = (nTiles + 3) / 4;    // 4 waves / 128-thread block
            gemm_wmma_f32<4><<<blocks, 128, 0, stream>>>(A, lda, W, ldw, bias,
                                                         C, ldc, M, Nn, K, act);
        } else {
            int nTiles = (M / 16) * (Nn / 16);
            int blocks = (nTiles + 3) / 4;
            gemm_wmma_f32<1><<<blocks, 128, 0, stream>>>(A, lda, W, ldw, bias,
                                                         C, ldc, M, Nn, K, act);
        }
    };
    auto gsz = [](int n) { return (n + 255) / 256; };

    // ---- pad weights ----
    pad_copy_kernel<<<gsz(144 * 256), 256, 0, stream>>>(iW0, 130, 256, Wp144, 144, 256);
    pad_copy_kernel<<<gsz(256 * 16), 256, 0, stream>>>(iW2, 256, 8, WpI2, 256, 16);
    pad_copy_kernel<<<gsz(16), 256, 0, stream>>>(ib2, 1, 8, bpI2, 1, 16);
    pad_copy_kernel<<<gsz(256 * 16), 256, 0, stream>>>(uW2, 256, 12, WpU2, 256, 16);
    pad_copy_kernel<<<gsz(16), 256, 0, stream>>>(ub2, 1, 12, bpU2, 1, 16);

    // ---- init_proj ----
    build_x0_kernel<<<gsz(NTOK * 144), 256, 0, stream>>>(h, X);
    ge(X, 144, Wp144, 256, ib0, H1, 256, NTOK, 256, 144, 1);
    ge(H1, 256, iW1, 256, ib1, H2, 256, NTOK, 256, 256, 1);
    ge(H2, 256, WpI2, 16, bpI2, U16, 16, NTOK, 16, 256, 0);
    equil_init_kernel<<<BB, NN, 0, stream>>>(U16, q, m, muq, mum);

    // ---- refinement iterations ----
    for (int it = 0; it < 2; ++it) {
        field_kernel<<<dim3(NN / 256, BB), 256, 0, stream>>>(pos, q, m, muq, mum,
                                                             vq, vm, eqb, emb);
        build_xu_kernel<<<gsz(NTOK * 144), 256, 0, stream>>>(h, q, m, muq, mum,
                                                             vq, vm, eqb, emb, X);
        ge(X, 144, uW0, 256, ub0, H1, 256, NTOK, 256, 144, 1);
        ge(H1, 256, uW1, 256, ub1, H2, 256, NTOK, 256, 256, 1);
        ge(H2, 256, WpU2, 16, bpU2, U16, 16, NTOK, 16, 256, 0);
        update_equil_kernel<<<BB, NN, 0, stream>>>(U16, q, m, muq, mum);
    }

    // ---- final field ----
    field_kernel<<<dim3(NN / 256, BB), 256, 0, stream>>>(pos, q, m, muq, mum,
                                                         vq, vm, eqb, emb);

    // ---- atom_sum MLP -> d_out ----
    build_atom_in_kernel<<<gsz(NTOK * 16), 256, 0, stream>>>(q, m, muq, mum,
                                                             vq, vm, eqb, emb, U16);
    ge(U16, 16, aW0, 256, ab0, H1, 256, NTOK, 256, 16, 1);
    ge(H1, 256, aW1, 256, ab1, H2, 256, NTOK, 256, 256, 1);
    ge(H2, 256, aW2, 256, ab2, out_atom, 256, NTOK, 256, 256, 0);

    // ---- abs_sum MLP (16 rows) -> d_out ----
    build_abs_in_kernel<<<1, 256, 0, stream>>>(q, m, muq, mum, vq, vm, eqb, emb,
                                               absIdx, absin);
    ge(absin, 16, sW0, 256, sb0, H1, 256, BB, 256, 16, 1);
    ge(H1, 256, sW1, 256, sb1, H2, 256, BB, 256, 256, 1);
    ge(H2, 256, sW2, 256, sb2, out_abs, 256, BB, 256, 256, 0);

    // ---- scalar/vector outputs ----
    copy_outputs_kernel<<<gsz(NTOK * 16), 256, 0, stream>>>(q, m, muq, mum,
                                                            vq, vm, eqb, emb, out);
}